// ThinFilm_25829933318282
// MI455X (gfx1250) — compile-verified
//
#include <hip/hip_runtime.h>
#include <stdint.h>

#define TPB    256            // threads per block
#define VPT    4              // outputs per thread
#define TILE   (TPB * VPT)    // 1024 outputs per block
#define HALO_L 4              // left halo (4 keeps the staged region 16B-aligned)
#define HALO_R 2
#define SLOTS  (TILE + HALO_L + HALO_R)   // 1030 staged h values, slot k <-> h[base-HALO_L+k]

// CDNA5 async copies: memory -> LDS with no VGPR round trip, tracked by ASYNCcnt.
__device__ __forceinline__ void async_ld_b32(uint32_t lds_byte, uint32_t voff_byte,
                                             uint64_t sbase) {
    asm volatile("global_load_async_to_lds_b32 %0, %1, %2"
                 :: "v"(lds_byte), "v"(voff_byte), "s"(sbase) : "memory");
}
__device__ __forceinline__ void async_ld_b128(uint32_t lds_byte, uint32_t voff_byte,
                                              uint64_t sbase) {
    asm volatile("global_load_async_to_lds_b128 %0, %1, %2"
                 :: "v"(lds_byte), "v"(voff_byte), "s"(sbase) : "memory");
}
__device__ __forceinline__ void wait_async_zero() {
    asm volatile("s_wait_asynccnt 0" ::: "memory");
}

__global__ __launch_bounds__(TPB) void ThinFilm_dhdt_kernel(
    const float* __restrict__ h,
    const float* __restrict__ pdx,
    const float* __restrict__ pQ,
    const float* __restrict__ ph0,
    float* __restrict__ out,
    int n)
{
    __shared__ float sm[TILE + 12];   // 1036 floats; fast path writes slots 0..1031

    const int t    = (int)threadIdx.x;
    const int base = (int)blockIdx.x * TILE;
    const uint64_t hbase  = (uint64_t)(uintptr_t)h;
    const uint32_t smbase = (uint32_t)(uintptr_t)&sm[0]; // low 32 bits of generic ptr == LDS offset

    // Edge blocks (first/last) need clamped addresses + BC substitution; all others
    // take the aligned wide-load fast path with no clamping at all.
    const bool edge = (base < HALO_L) || (base + TILE + HALO_L > n);

    if (!edge) {
        // staged idx range: [base-4, base+1027], all in [0, n-1] by the edge test
        const uint64_t sb = hbase + (uint64_t)(uint32_t)((base - HALO_L) * 4);
        async_ld_b128(smbase + 16u * (uint32_t)t, 16u * (uint32_t)t, sb);     // slots 4t..4t+3
        if (t < 2)
            async_ld_b128(smbase + 4096u + 16u * (uint32_t)t,
                          4096u + 16u * (uint32_t)t, sb);                     // slots 1024..1031
    } else {
        const int cap = n - 2;   // clamp to n-2: outlet BC h[n-1] = h[n-2] comes for free
#pragma unroll
        for (int r = 0; r < VPT; ++r) {
            int k   = t + r * TPB;            // slots 0..1023
            int idx = base - HALO_L + k;
            idx = idx < 0 ? 0 : idx;
            idx = idx > cap ? cap : idx;
            async_ld_b32(smbase + 4u * (uint32_t)k, (uint32_t)(idx * 4), hbase);
        }
        if (t < (HALO_L + HALO_R)) {          // slots 1024..1029
            int k   = TILE + t;
            int idx = base - HALO_L + k;
            idx = idx < 0 ? 0 : idx;
            idx = idx > cap ? cap : idx;
            async_ld_b32(smbase + 4u * (uint32_t)k, (uint32_t)(idx * 4), hbase);
        }
    }

    // uniform scalar parameters (SMEM loads) — issue while the async copies fly
    const float dx  = *pdx;
    const float Q   = *pQ;
    const float h0v = *ph0;

    wait_async_zero();   // this wave's async copies have landed in LDS
    __syncthreads();     // all waves' copies visible

    if (base == 0) {     // inlet BC: h[0] := h0 (slot HALO_L holds global index 0)
        if (t == 0) sm[HALO_L] = h0v;
        __syncthreads();
    }

    const float inv_dx  = 1.0f / dx;              // dx = 1.0 in the test: exact
    const float inv_dx3 = inv_dx * inv_dx * inv_dx;

    // hv[k] <-> h[i0 - 4 + k]; reads merge into ds_load_b128 x2 + ds_load_b64
    float hv[10];
#pragma unroll
    for (int k = 0; k < 10; ++k) hv[k] = sm[VPT * t + k];

    const int i0 = base + VPT * t;
    float res[VPT];
#pragma unroll
    for (int j = 0; j < VPT; ++j) {
        const int i = i0 + j;
        const float hm3 = hv[j + 1];  // h[i-3] (used only at i == n-1)
        const float hm2 = hv[j + 2];
        const float hm1 = hv[j + 3];
        const float hc  = hv[j + 4];
        const float hp1 = hv[j + 5];
        const float hp2 = hv[j + 6];

        // q(j) = (0.5*(h[j+1]+h[j+2]))^3 * (h[j+3]-3h[j+2]+3h[j+1]-h[j]) / dx^3
        const float ar  = 0.5f * (hc + hp1);                               // right face q(i-1)
        const float d3r = (hp2 - 3.0f * hp1 + 3.0f * hc - hm1) * inv_dx3;
        const float q_r = ar * ar * ar * d3r;
        const float al  = 0.5f * (hm1 + hc);                               // left face q(i-2)
        const float d3l = (hp1 - 3.0f * hc + 3.0f * hm1 - hm2) * inv_dx3;
        const float q_l = al * al * al * d3l;

        float r;
        if (i == 0) {
            r = 0.0f;
        } else if (i == 1) {
            r = (Q - q_r) * inv_dx;               // inlet node: prescribed flux
        } else if (i <= n - 3) {
            r = (q_l - q_r) * inv_dx;             // interior
        } else if (i == n - 2) {
            r = (q_l - hc) * inv_dx;              // outlet node: q_l == q(n-4)
        } else {                                  // i == n-1: recompute q(n-4) = q(i-3)
            const float all_ = 0.5f * (hm2 + hm1);
            const float d3ll = (hc - 3.0f * hm1 + 3.0f * hm2 - hm3) * inv_dx3;
            const float q_ll = all_ * all_ * all_ * d3ll;
            r = (q_ll - hm1) * inv_dx;            // hm1 == h[n-2]
        }
        res[j] = r;
    }

    if (i0 + VPT <= n) {
        float4 v4;
        v4.x = res[0]; v4.y = res[1]; v4.z = res[2]; v4.w = res[3];
        *(float4*)(out + i0) = v4;                // 16B-aligned b128 store
    } else {
#pragma unroll
        for (int j = 0; j < VPT; ++j)
            if (i0 + j < n) out[i0 + j] = res[j];
    }
}

extern "C" void kernel_launch(void* const* d_in, const int* in_sizes, int n_in,
                              void* d_out, int out_size, void* d_ws, size_t ws_size,
                              hipStream_t stream) {
    // setup_inputs order: t(int), h(float[N]), dx, Q, h0 (each float[1])
    const float* h  = (const float*)d_in[1];
    const float* dx = (const float*)d_in[2];
    const float* Q  = (const float*)d_in[3];
    const float* h0 = (const float*)d_in[4];
    float* out = (float*)d_out;
    const int n = in_sizes[1];
    const int blocks = (n + TILE - 1) / TILE;
    ThinFilm_dhdt_kernel<<<blocks, TPB, 0, stream>>>(h, dx, Q, h0, out, n);
}